// Joiner_56075093017101
// MI455X (gfx1250) — compile-verified
//
#include <hip/hip_runtime.h>

// Problem dims (fixed by the reference)
constexpr int Bn = 8;
constexpr int Tn = 256;
constexpr int Un = 64;
constexpr int Hn = 1024;
constexpr int Vn = 128;

typedef __attribute__((ext_vector_type(2))) float v2f;
typedef __attribute__((ext_vector_type(4))) float v4f;
typedef __attribute__((ext_vector_type(8))) float v8f;

// -------------------------------------------------------------------------
// Stage 1: E[bt, v] = enc[bt,:] . W[v,:] + bias[v]   (2048 x 128, K=1024)
//          D[bu, v] = dec[bu,:] . W[v,:]             ( 512 x 128, K=1024)
// One wave32 per 16x16 fp32 output tile using V_WMMA_F32_16X16X4_F32.
//
// A layout (ISA 7.12.2, 32-bit A 16x4): lane l<16 -> M=l, holds K=0,1 in
// vgpr0/1; lanes 16-31 -> M=l-16, K=2,3.  B mirrors with N=lane%16.  Since
// B[k][n] = W[n][k], both A and B lane loads are contiguous float2 pairs
// from row-major enc/dec/W.
// -------------------------------------------------------------------------
constexpr int E_MT = (Bn * Tn) / 16;   // 128 M-tiles for E
constexpr int D_MT = (Bn * Un) / 16;   //  32 M-tiles for D
constexpr int NT   = Vn / 16;          //   8 N-tiles
constexpr int E_TILES = E_MT * NT;     // 1024
constexpr int D_TILES = D_MT * NT;     //  256
constexpr int ALL_TILES = E_TILES + D_TILES;  // 1280
constexpr int WAVES_PER_BLOCK = 4;

__global__ __launch_bounds__(32 * WAVES_PER_BLOCK)
void joiner_gemm_wmma(const float* __restrict__ enc,
                      const float* __restrict__ dec,
                      const float* __restrict__ Wm,
                      const float* __restrict__ bias,
                      float* __restrict__ E,
                      float* __restrict__ D)
{
    int tile = blockIdx.x * WAVES_PER_BLOCK + (threadIdx.x >> 5);
    const int lane = threadIdx.x & 31;

    const float* src;
    float*       dst;
    float        bscale;
    if (tile < E_TILES) {
        src = enc; dst = E; bscale = 1.0f;
    } else {
        tile -= E_TILES;
        src = dec; dst = D; bscale = 0.0f;
    }

    const int tm   = tile >> 3;        // M tile
    const int tn   = tile & 7;         // N (V) tile
    const int row0 = tm * 16;
    const int col0 = tn * 16;

    const int half = lane >> 4;        // 0: K=0,1  1: K=2,3
    const int q    = lane & 15;        // M index for A, N index for B

    const float* aptr = src + (size_t)(row0 + q) * Hn + 2 * half;
    const float* bptr = Wm  + (size_t)(col0 + q) * Hn + 2 * half;

    v8f acc = {};
#pragma unroll 8
    for (int k = 0; k < Hn; k += 4) {
        v2f a = *reinterpret_cast<const v2f*>(aptr + k);
        v2f b = *reinterpret_cast<const v2f*>(bptr + k);
        // D = A(16x4) x B(4x16) + C  -> v_wmma_f32_16x16x4_f32
        acc = __builtin_amdgcn_wmma_f32_16x16x4_f32(
            /*neg_a=*/false, a, /*neg_b=*/false, b,
            /*c_mod=*/(short)0, acc, /*reuse_a=*/false, /*reuse_b=*/false);
    }

    // C/D layout: lane -> n = lane%16 ; vgpr r -> m = r + 8*(lane/16)
    const float bb = bscale * bias[col0 + q];
#pragma unroll
    for (int r = 0; r < 8; ++r) {
        dst[(size_t)(row0 + r + 8 * half) * Vn + col0 + q] = acc[r] + bb;
    }
}

// -------------------------------------------------------------------------
// Stage 2: out[bt,u,v] = softmax_v( E[bt,v] + D[b*U+u,v] )
// One block per (b,t); 8 waves, each wave handles 8 u-rows.
// V=128 -> 4 floats per lane; wave32 shuffle reductions for max & sum.
// Non-temporal stores keep the 64MB output from evicting E/D out of L2.
// -------------------------------------------------------------------------
__device__ __forceinline__ float waveMax32(float v) {
#pragma unroll
    for (int off = 16; off > 0; off >>= 1)
        v = fmaxf(v, __shfl_xor(v, off, 32));
    return v;
}
__device__ __forceinline__ float waveSum32(float v) {
#pragma unroll
    for (int off = 16; off > 0; off >>= 1)
        v += __shfl_xor(v, off, 32);
    return v;
}

__global__ __launch_bounds__(256)
void joiner_softmax(const float* __restrict__ E,
                    const float* __restrict__ D,
                    float* __restrict__ out)
{
    const int bt   = blockIdx.x;          // 0 .. B*T-1
    const int b    = bt / Tn;
    const int lane = threadIdx.x & 31;
    const int wave = threadIdx.x >> 5;    // 0..7

    const v4f e = *reinterpret_cast<const v4f*>(E + (size_t)bt * Vn + lane * 4);
    const float* drow = D   + (size_t)b  * Un * Vn + lane * 4;
    float*       orow = out + (size_t)bt * Un * Vn + lane * 4;

    constexpr float L2E = 1.4426950408889634f;

    for (int u = wave; u < Un; u += 8) {
        v4f d = *reinterpret_cast<const v4f*>(drow + (size_t)u * Vn);
        v4f x = e + d;

        float m = fmaxf(fmaxf(x.x, x.y), fmaxf(x.z, x.w));
        m = waveMax32(m);

        v4f p;
        p.x = __builtin_amdgcn_exp2f((x.x - m) * L2E);
        p.y = __builtin_amdgcn_exp2f((x.y - m) * L2E);
        p.z = __builtin_amdgcn_exp2f((x.z - m) * L2E);
        p.w = __builtin_amdgcn_exp2f((x.w - m) * L2E);

        float s = p.x + p.y + p.z + p.w;
        s = waveSum32(s);
        const float inv = 1.0f / s;

        v4f o = p * inv;
        __builtin_nontemporal_store(o, reinterpret_cast<v4f*>(orow + (size_t)u * Vn));
    }
}

// -------------------------------------------------------------------------
// Host side
// -------------------------------------------------------------------------
extern "C" void kernel_launch(void* const* d_in, const int* in_sizes, int n_in,
                              void* d_out, int out_size, void* d_ws, size_t ws_size,
                              hipStream_t stream)
{
    const float* enc  = (const float*)d_in[0];  // [B,T,H]
    const float* dec  = (const float*)d_in[1];  // [B,U,H]
    const float* Wm   = (const float*)d_in[2];  // [V,H]
    const float* bias = (const float*)d_in[3];  // [V]
    float* out = (float*)d_out;                 // [B,T,U,V]

    float* E = (float*)d_ws;                    // [B*T, V]  (1 MB)
    float* D = E + (size_t)Bn * Tn * Vn;        // [B*U, V]  (256 KB)

    // Stage 1: 1280 wave-tiles, 4 waves per 128-thread block -> 320 blocks
    joiner_gemm_wmma<<<ALL_TILES / WAVES_PER_BLOCK, 32 * WAVES_PER_BLOCK, 0, stream>>>(
        enc, dec, Wm, bias, E, D);

    // Stage 2: one block per (b,t)
    joiner_softmax<<<Bn * Tn, 256, 0, stream>>>(E, D, out);
}